// TanhNewtonLayer_1219770712655
// MI455X (gfx1250) — compile-verified
//
#include <hip/hip_runtime.h>

#define B_ROWS   4096
#define D_DIM    64
#define ROWS_PB  16
#define THREADS  256
#define NWAVE    8
#define MAX_NEWTON 50
#define TOL2   (1e-4f * 1e-4f)   // final per-row zeroing test (norm > 1e-4)
#define STOP2  (4e-12f)          // per-row Newton stop: ||g|| <= 2e-6

typedef __attribute__((ext_vector_type(2))) float v2f;
typedef __attribute__((ext_vector_type(8))) float v8f;

__global__ __launch_bounds__(THREADS, 1)
void tanh_newton_kernel(const float* __restrict__ xg,
                        const float* __restrict__ Wg,
                        float* __restrict__ out)
{
    // LDS: 8 LU workspaces (64x65 each) + z^T + zl + g + per-row residuals
    __shared__ float sA[NWAVE * D_DIM * (D_DIM + 1)]; // 133,120 B
    __shared__ float sZ[D_DIM * ROWS_PB];             // z transposed: [k][r]
    __shared__ float sZL[ROWS_PB * D_DIM];            // zl row-major
    __shared__ float sG[ROWS_PB * D_DIM];             // residual row-major
    __shared__ float sRes[ROWS_PB];

    const int t    = threadIdx.x;
    const int w    = t >> 5;        // wave id 0..7
    const int l    = t & 31;        // lane id
    const int row0 = blockIdx.x * ROWS_PB;

    const int n16 = l & 15;             // M for A-operand, N for B-operand
    const int hiK = (l >> 4) << 1;      // hi half-wave holds K+2
    const int hiM = (l >> 4) << 3;      // hi half-wave holds M+8 in C/D

    // ---- init z0 = tanh(x) ----
    for (int e = t; e < ROWS_PB * D_DIM; e += THREADS) {
        int r = e >> 6, c = e & 63;
        sZ[c * ROWS_PB + r] = tanhf(xg[(row0 + r) * D_DIM + c]);
    }

    // ---- preload WMMA B-operand (W columns) into registers: W never changes ----
    // B(k,n) must equal W[n][k]  (zl[m][n] = sum_k z[m][k] * W[n][k])
    float bx[16], by[16];
    if (w < 4) {
        const int ncol = w * 16 + n16;
        #pragma unroll
        for (int kt = 0; kt < 16; ++kt) {
            bx[kt] = Wg[ncol * D_DIM + 4 * kt + hiK];
            by[kt] = Wg[ncol * D_DIM + 4 * kt + 1 + hiK];
        }
    }
    __syncthreads();

    for (int it = 0; ; ++it) {
        // ---- zl = z @ W^T + x via V_WMMA_F32_16X16X4_F32 (waves 0..3) ----
        if (w < 4) {
            v8f acc = {};
            #pragma unroll
            for (int kt = 0; kt < 16; ++kt) {
                const int k0 = 4 * kt;
                v2f a, b;
                a.x = sZ[(k0 + hiK) * ROWS_PB + n16];       // A: M=n16, K=k0+hiK
                a.y = sZ[(k0 + 1 + hiK) * ROWS_PB + n16];   //            k0+1+hiK
                b.x = bx[kt];
                b.y = by[kt];
                acc = __builtin_amdgcn_wmma_f32_16x16x4_f32(
                          false, a, false, b, (short)0, acc, false, false);
            }
            const int ncol = w * 16 + n16;
            #pragma unroll
            for (int v = 0; v < 8; ++v) {
                const int mrow = v + hiM;
                sZL[mrow * D_DIM + ncol] =
                    acc[v] + xg[(row0 + mrow) * D_DIM + ncol];
            }
        }
        __syncthreads();

        // ---- residual g = z - tanh(zl) ----
        for (int e = t; e < ROWS_PB * D_DIM; e += THREADS) {
            int r = e >> 6, c = e & 63;
            sG[e] = sZ[c * ROWS_PB + r] - tanhf(sZL[e]);
        }
        __syncthreads();

        // ---- deterministic per-row ||g||^2 ----
        if (t < ROWS_PB) {
            float acc = 0.0f;
            for (int c = 0; c < D_DIM; ++c) {
                float g = sG[t * D_DIM + c];
                acc += g * g;
            }
            sRes[t] = acc;
        }
        __syncthreads();

        bool done = (it >= MAX_NEWTON);
        if (!done) {
            done = true;
            #pragma unroll
            for (int r = 0; r < ROWS_PB; ++r)
                if (sRes[r] > STOP2) done = false;
        }
        __syncthreads();   // separates sRes reads from next-iteration writes
        if (done) break;

        // ---- per-wave LU solves: wave w handles batch rows w and w+8 ----
        float* A = &sA[w * D_DIM * (D_DIM + 1)];
        #pragma unroll 1
        for (int ro = 0; ro < 2; ++ro) {
            const int r = w + 8 * ro;

            // build [ J | g ] ; J[i][j] = delta_ij - sech^2(zl_i)*W[i][j]
            #pragma unroll 1
            for (int half = 0; half < 2; ++half) {
                const int i  = l + 32 * half;
                const float th = sZ[i * ROWS_PB + r] - sG[r * D_DIM + i]; // tanh(zl_i)
                const float si = 1.0f - th * th;                          // sech^2
                const float4* W4 = reinterpret_cast<const float4*>(Wg + i * D_DIM);
                #pragma unroll
                for (int j4 = 0; j4 < 16; ++j4) {
                    float4 wv = W4[j4];
                    int j = 4 * j4;
                    A[i * 65 + j + 0] = ((i == j + 0) ? 1.0f : 0.0f) - si * wv.x;
                    A[i * 65 + j + 1] = ((i == j + 1) ? 1.0f : 0.0f) - si * wv.y;
                    A[i * 65 + j + 2] = ((i == j + 2) ? 1.0f : 0.0f) - si * wv.z;
                    A[i * 65 + j + 3] = ((i == j + 3) ? 1.0f : 0.0f) - si * wv.w;
                }
                A[i * 65 + 64] = sG[r * D_DIM + i];
            }

            // forward elimination with partial pivoting (wave-local, no barriers)
            #pragma unroll 1
            for (int k = 0; k < D_DIM; ++k) {
                // pivot search: lane candidates are its owned rows l and l+32
                float bv = -1.0f; int bi = k;
                if (l >= k) {
                    float vv = fabsf(A[l * 65 + k]);
                    if (vv > bv) { bv = vv; bi = l; }
                }
                {
                    int i2 = l + 32;
                    float vv = fabsf(A[i2 * 65 + k]);
                    if (i2 >= k && (vv > bv || (vv == bv && i2 < bi))) { bv = vv; bi = i2; }
                }
                #pragma unroll
                for (int off = 16; off > 0; off >>= 1) {
                    float ov = __shfl_xor(bv, off, 32);
                    int   oi = __shfl_xor(bi, off, 32);
                    if (ov > bv || (ov == bv && oi < bi)) { bv = ov; bi = oi; }
                }
                if (bi != k) {   // uniform within wave: column-parallel row swap
                    float t0 = A[k * 65 + l];        A[k * 65 + l]        = A[bi * 65 + l];        A[bi * 65 + l]        = t0;
                    float t1 = A[k * 65 + l + 32];   A[k * 65 + l + 32]   = A[bi * 65 + l + 32];   A[bi * 65 + l + 32]   = t1;
                    if (l == 0) {
                        float t2 = A[k * 65 + 64];   A[k * 65 + 64]       = A[bi * 65 + 64];       A[bi * 65 + 64]       = t2;
                    }
                }
                const float invp = 1.0f / A[k * 65 + k];
                const int  i1 = l + 32;
                const bool d0 = (l  > k);
                const bool d1 = (i1 > k);
                const float f0 = d0 ? A[l  * 65 + k] * invp : 0.0f;
                const float f1 = d1 ? A[i1 * 65 + k] * invp : 0.0f;
                for (int j = k + 1; j <= D_DIM; ++j) {
                    const float pk = A[k * 65 + j];   // broadcast read
                    if (d0) A[l  * 65 + j] -= f0 * pk;
                    if (d1) A[i1 * 65 + j] -= f1 * pk;
                }
            }

            // back substitution: solution accumulates in the augmented column
            #pragma unroll 1
            for (int k = D_DIM - 1; k >= 0; --k) {
                const float xk = A[k * 65 + 64] / A[k * 65 + k]; // broadcast
                if (l < k)        A[l        * 65 + 64] -= A[l        * 65 + k] * xk;
                if (l + 32 < k)   A[(l + 32) * 65 + 64] -= A[(l + 32) * 65 + k] * xk;
                if (l == 0)       A[k * 65 + 64] = xk;
            }

            // z <- z - dz  (column r of sZ; no cross-wave overlap)
            sZ[l        * ROWS_PB + r] -= A[l        * 65 + 64];
            sZ[(l + 32) * ROWS_PB + r] -= A[(l + 32) * 65 + 64];
        }
        __syncthreads();
    }

    // ---- output: zero rows with ||g|| > TOL ----
    for (int e = t; e < ROWS_PB * D_DIM; e += THREADS) {
        int r = e >> 6, c = e & 63;
        float zv = sZ[c * ROWS_PB + r];
        out[(row0 + r) * D_DIM + c] = (sRes[r] > TOL2) ? 0.0f : zv;
    }
}

extern "C" void kernel_launch(void* const* d_in, const int* in_sizes, int n_in,
                              void* d_out, int out_size, void* d_ws, size_t ws_size,
                              hipStream_t stream) {
    (void)in_sizes; (void)n_in; (void)out_size; (void)d_ws; (void)ws_size;
    const float* x = (const float*)d_in[0];   // [4096, 64]
    const float* W = (const float*)d_in[1];   // [64, 64]
    float* out = (float*)d_out;               // [4096, 64]
    tanh_newton_kernel<<<dim3(B_ROWS / ROWS_PB), dim3(THREADS), 0, stream>>>(x, W, out);
}